// Histeq_39436389711987
// MI455X (gfx1250) — compile-verified
//
#include <hip/hip_runtime.h>

typedef float v2f __attribute__((ext_vector_type(2)));
typedef float v8f __attribute__((ext_vector_type(8)));
typedef float f4  __attribute__((ext_vector_type(4)));
typedef int   i4  __attribute__((ext_vector_type(4)));

#define NBINS 256
#define NCH   192                 // 64 images * 3 channels
#define HW    (512 * 512)         // 262144 pixels per channel
#define BPC   8                   // blocks per channel (streaming passes)
#define TPB   256                 // threads per block = 8 waves (wave32)
#define CHUNK (HW / BPC)          // 32768 pixels per block
#define VITER (CHUNK / (TPB * 4)) // 32 int4 iterations per thread

// ---------------------------------------------------------------- zero hist
__global__ void histeq_zero_kernel(int* __restrict__ ghist) {
    ghist[blockIdx.x * TPB + threadIdx.x] = 0;
}

// ---------------------------------------------------------------- pass 1: histogram
// Per-wave private LDS sub-histograms (ds_add_u32, conflicts only within one
// wave32), b128 coalesced loads with default (RT) temporal hint so the input
// stays resident in the 192MB L2 for the apply pass.
__global__ void histeq_hist_kernel(const int* __restrict__ pic,
                                   int* __restrict__ ghist) {
    __shared__ int lh[8 * NBINS];                 // 8 waves * 256 bins = 8KB
    const int tid = threadIdx.x;
    for (int i = tid; i < 8 * NBINS; i += TPB) lh[i] = 0;
    __syncthreads();

    const int ch  = blockIdx.x / BPC;
    const int sub = blockIdx.x % BPC;
    const i4* src = reinterpret_cast<const i4*>(pic + (size_t)ch * HW + (size_t)sub * CHUNK);

    int* myh = lh + (tid >> 5) * NBINS;           // wave32-private histogram
    for (int it = 0; it < VITER; ++it) {
        i4 v = src[it * TPB + tid];
        atomicAdd(&myh[v.x & 255], 1);
        atomicAdd(&myh[v.y & 255], 1);
        atomicAdd(&myh[v.z & 255], 1);
        atomicAdd(&myh[v.w & 255], 1);
    }
    __syncthreads();

    int s = 0;
    for (int w = 0; w < 8; ++w) s += lh[w * NBINS + tid];
    atomicAdd(&ghist[ch * NBINS + tid], s);
}

// ---------------------------------------------------------------- pass 2: LUT via WMMA
// One wave32 per channel. hist viewed as 16x16 matrix H (H[r][c] = h[16r+c]).
// CDF = H*T + L*W  with T upper-tri(ones incl diag), L strict-lower-tri(ones),
// W[k][n] = rowsum[k]. Two 16x16x16 products = 8x v_wmma_f32_16x16x4_f32.
// All counts <= 262144 < 2^24 -> exact in f32.
__global__ void histeq_lut_kernel(const int* __restrict__ ghist,
                                  int* __restrict__ lut,
                                  int* __restrict__ gstep) {
    __shared__ int   sh[NBINS];
    __shared__ float rowsum[16];
    __shared__ int   s_safe;

    const int lane = threadIdx.x;      // 0..31
    const int ch   = blockIdx.x;
    const int* h   = ghist + ch * NBINS;

    for (int i = 0; i < 8; ++i) sh[lane + 32 * i] = h[lane + 32 * i];
    __syncthreads();

    if (lane < 16) {
        float s = 0.f;
        for (int c = 0; c < 16; ++c) s += (float)sh[lane * 16 + c];
        rowsum[lane] = s;
    }
    __syncthreads();

    const int m  = lane & 15;          // A-row / B-col / D-col index
    const int hi = lane >> 4;          // half-select (K for A/B, M+8 for D)

    v8f acc = {};
    // ---- H * T : within-row prefix sums
    for (int q = 0; q < 4; ++q) {
        const int k0 = 4 * q + 2 * hi; // K indices k0, k0+1 for this lane half
        v2f a, b;
        a.x = (float)sh[16 * m + k0];
        a.y = (float)sh[16 * m + k0 + 1];
        b.x = (k0     <= m) ? 1.f : 0.f;   // T[k][n], n == lane&15
        b.y = (k0 + 1 <= m) ? 1.f : 0.f;
        acc = __builtin_amdgcn_wmma_f32_16x16x4_f32(false, a, false, b,
                                                    (short)0, acc, false, false);
    }
    // ---- + L * W : add prefix of full-row sums
    for (int q = 0; q < 4; ++q) {
        const int k0 = 4 * q + 2 * hi;
        v2f a, b;
        a.x = (k0     < m) ? 1.f : 0.f;    // L[m][k] strict lower
        a.y = (k0 + 1 < m) ? 1.f : 0.f;
        b.x = rowsum[k0];                  // W[k][n] = rowsum[k] (bcast over n)
        b.y = rowsum[k0 + 1];
        acc = __builtin_amdgcn_wmma_f32_16x16x4_f32(false, a, false, b,
                                                    (short)0, acc, false, false);
    }

    // last nonzero bin, step, safe_step (trivial scalar work, lane 0)
    if (lane == 0) {
        int last = NBINS - 1;
        while (last > 0 && sh[last] == 0) --last;
        int step = (HW - sh[last]) / (NBINS - 1);
        gstep[ch] = step;
        s_safe    = step < 1 ? 1 : step;
        lut[ch * NBINS] = 0;               // lutF[0] = clip(0) = 0
    }
    __syncthreads();

    const int safe = s_safe;
    const int hstep = safe >> 1;           // safe_step // 2
    // D layout: vgpr v, lane -> element (M = v + 8*hi, N = lane&15)
    for (int v = 0; v < 8; ++v) {
        const int i = 16 * (v + 8 * hi) + m;   // cdf index this lane holds
        if (i < NBINS - 1) {                   // cdf[i] feeds lutF[i+1]
            int cdf = (int)(acc[v] + 0.5f);    // exact integer in f32
            int val = (cdf + hstep) / safe;
            if (val > 255) val = 255;
            lut[ch * NBINS + i + 1] = val;
        }
    }
}

// ---------------------------------------------------------------- pass 3: apply LUT
// LUT gathered from LDS; non-temporal loads (input is dead after this) and
// non-temporal stores (output never re-read) to keep L2 clean.
__global__ void histeq_apply_kernel(const int* __restrict__ pic,
                                    const int* __restrict__ lut,
                                    const int* __restrict__ gstep,
                                    float* __restrict__ out) {
    __shared__ int slut[NBINS];
    __shared__ int s_step;
    const int tid = threadIdx.x;
    const int ch  = blockIdx.x / BPC;
    const int sub = blockIdx.x % BPC;

    slut[tid] = lut[ch * NBINS + tid];
    if (tid == 0) s_step = gstep[ch];
    __syncthreads();

    const size_t base = (size_t)ch * HW + (size_t)sub * CHUNK;
    const i4* src = reinterpret_cast<const i4*>(pic + base);
    f4*       dst = reinterpret_cast<f4*>(out + base);
    const int step = s_step;

    for (int it = 0; it < VITER; ++it) {
        const int idx = it * TPB + tid;
        i4 v = __builtin_nontemporal_load(&src[idx]);
        f4 r;
        if (step == 0) {   // torchvision: channel returned unchanged
            r.x = (float)v.x; r.y = (float)v.y; r.z = (float)v.z; r.w = (float)v.w;
        } else {
            r.x = (float)slut[v.x & 255];
            r.y = (float)slut[v.y & 255];
            r.z = (float)slut[v.z & 255];
            r.w = (float)slut[v.w & 255];
        }
        __builtin_nontemporal_store(r, &dst[idx]);
    }
}

// ---------------------------------------------------------------- launcher
extern "C" void kernel_launch(void* const* d_in, const int* in_sizes, int n_in,
                              void* d_out, int out_size, void* d_ws, size_t ws_size,
                              hipStream_t stream) {
    const int* pic = (const int*)d_in[0];
    float*     out = (float*)d_out;

    int* ghist = (int*)d_ws;               // 192*256 ints
    int* lutb  = ghist + NCH * NBINS;      // 192*256 ints
    int* gstep = lutb  + NCH * NBINS;      // 192 ints

    histeq_zero_kernel <<<NCH,       TPB, 0, stream>>>(ghist);
    histeq_hist_kernel <<<NCH * BPC, TPB, 0, stream>>>(pic, ghist);
    histeq_lut_kernel  <<<NCH,       32,  0, stream>>>(ghist, lutb, gstep);
    histeq_apply_kernel<<<NCH * BPC, TPB, 0, stream>>>(pic, lutb, gstep, out);
}